// Decode_76244259439060
// MI455X (gfx1250) — compile-verified
//
#include <hip/hip_runtime.h>
#include <hip/hip_bf16.h>
#include <math.h>

typedef float v2f __attribute__((ext_vector_type(2)));
typedef float v4f __attribute__((ext_vector_type(4)));
typedef float v8f __attribute__((ext_vector_type(8)));

namespace {
constexpr int B_    = 512;
constexpr int HQK_  = 4;
constexpr int HV_   = 8;
constexpr int D_    = 128;
constexpr int TS    = 132;               // padded LDS row stride (floats): bank-conflict-free A gather
constexpr int WAVES = 4;                 // 128-thread block, one (b,h) per block
constexpr float SCALE = 0.08838834764831843f;  // 1/sqrt(128)
}

__global__ __launch_bounds__(WAVES * 32)
void gdn_decode_kernel(const float* __restrict__ q,
                       const float* __restrict__ k,
                       const float* __restrict__ v,
                       const float* __restrict__ state,
                       const float* __restrict__ A_log,
                       const float* __restrict__ a_param,
                       const float* __restrict__ dt_bias,
                       const float* __restrict__ b_param,
                       __hip_bfloat16* __restrict__ out,
                       float* __restrict__ new_state)
{
    __shared__ float sk[D_], sq[D_], sv[D_];
    __shared__ float s12[2][D_];                 // s1 / s2 per state row
    __shared__ float sg[2];                      // g, beta
    __shared__ float tile[WAVES][16 * TS];       // 4 x 8448 B padded state tiles

    const int bh   = blockIdx.x;
    const int b    = bh >> 3;                    // / HV_
    const int h    = bh & 7;                     // % HV_
    const int tid  = threadIdx.x;
    const int wave = tid >> 5;
    const int lane = tid & 31;

    // stage q/k (GQA head h>>1), v, and gate scalars
    if (tid < D_) {
        const int qh = h >> 1;
        sq[tid] = q[(b * HQK_ + qh) * D_ + tid];
        sk[tid] = k[(b * HQK_ + qh) * D_ + tid];
        sv[tid] = v[(b * HV_  + h ) * D_ + tid];
    }
    if (tid == 0) {
        float x  = a_param[b * HV_ + h] + dt_bias[h];
        float sp = (x > 0.f) ? (x + log1pf(expf(-x))) : log1pf(expf(x)); // softplus
        sg[0] = expf(-expf(A_log[h]) * sp);                              // g
        sg[1] = 1.f / (1.f + expf(-b_param[b * HV_ + h]));               // beta
    }
    __syncthreads();

    const float g    = sg[0];
    const float beta = sg[1];

    // qk = q·k (wave-local shuffle reduce; all lanes end with the sum)
    float qk = 0.f;
    for (int i = lane; i < D_; i += 32) qk += sk[i] * sq[i];
#pragma unroll
    for (int m = 16; m >= 1; m >>= 1) qk += __shfl_xor(qk, m, 32);

    const float* __restrict__ stbase = state     + (size_t)bh * D_ * D_;
    float*       __restrict__ nsbase = new_state + (size_t)bh * D_ * D_;

    // WMMA A-fragment addressing (16x4 f32): lanes 0-15 -> K={0,1}, lanes 16-31 -> K={2,3}
    const int   arow = lane & 15;
    const int   kq   = lane >> 4;
    const float* bs  = (arow == 0) ? sk : sq;     // B col 0 = k, col 1 = q
    const float  bm  = (arow < 2) ? 1.f : 0.f;    // zero all other B columns

    for (int t = 0; t < 2; ++t) {
        const int tbase = 16 * (wave + WAVES * t);   // first state row of this wave's tile

        // ---- async-copy 16 contiguous state rows (8 KB) into the padded LDS tile
        // (default RT temporal hint: input state is the only stream that can
        //  profit from the 192 MB L2 across calls)
        {
            const float*  gsrc    = stbase + (size_t)tbase * D_;
            const unsigned ldsbase = (unsigned)(size_t)(&tile[wave][0]);
#pragma unroll
            for (int i = 0; i < 16; ++i) {
                const int f  = i * 32 + lane;        // float4 index 0..511 in tile
                const int r  = f >> 5;               // row within tile
                const int c4 = f & 31;               // float4 column
                const unsigned laddr = ldsbase + (unsigned)(r * TS + c4 * 4) * 4u;
                const float*   gaddr = gsrc + (size_t)f * 4;
                asm volatile("global_load_async_to_lds_b128 %0, %1, off"
                             :: "v"(laddr), "v"(gaddr) : "memory");
            }
            asm volatile("s_wait_asynccnt 0x0" ::: "memory");
        }

        // ---- s1 = k·row, s2 = q·row for 16 rows via V_WMMA_F32_16X16X4_F32 over K chunks
        v8f acc = {0.f, 0.f, 0.f, 0.f, 0.f, 0.f, 0.f, 0.f};
        const float* rp = &tile[wave][arow * TS + kq * 2];
#pragma unroll 8
        for (int c = 0; c < D_; c += 4) {
            v2f a;  a[0]  = rp[c];               a[1]  = rp[c + 1];
            v2f bb; bb[0] = bs[c + 2 * kq] * bm; bb[1] = bs[c + 2 * kq + 1] * bm;
            acc = __builtin_amdgcn_wmma_f32_16x16x4_f32(false, a, false, bb,
                                                        (short)0, acc, false, false);
        }

        // C[m][0]=s1[m] (lane 0 / lane 16), C[m][1]=s2[m] (lane 1 / lane 17); scatter to LDS
        if (lane < 2 || (lane >= 16 && lane < 18)) {
            const int which = lane & 1;
            const int half  = (lane >> 4) * 8;
#pragma unroll
            for (int r = 0; r < 8; ++r) s12[which][tbase + half + r] = acc[r];
        }
        __syncthreads();

        // ---- row-local update: new_state = g*row + delta*k.
        // Output is dead data (never re-read on device): stream it past L2
        // with non-temporal stores so input-state lines stay resident.
#pragma unroll 4
        for (int r = 0; r < 16; ++r) {
            const int   row   = tbase + r;
            const float s1    = s12[0][row];
            const float s2    = s12[1][row];
            const float delta = beta * (sv[row] - g * s1);
            v4f rv = *(const v4f*)&tile[wave][r * TS + lane * 4];
            v4f k4 = *(const v4f*)&sk[lane * 4];
            v4f nr = g * rv + delta * k4;
            __builtin_nontemporal_store(nr, (v4f*)&nsbase[(size_t)row * D_ + lane * 4]);
            if (lane == 0) {
                __hip_bfloat16 ob = __float2bfloat16(SCALE * (g * s2 + qk * delta));
                __builtin_nontemporal_store(__builtin_bit_cast(unsigned short, ob),
                                            (unsigned short*)&out[bh * D_ + row]);
            }
        }
        __syncthreads();
    }
}

extern "C" void kernel_launch(void* const* d_in, const int* in_sizes, int n_in,
                              void* d_out, int out_size, void* d_ws, size_t ws_size,
                              hipStream_t stream) {
    const float* q       = (const float*)d_in[0];
    const float* k       = (const float*)d_in[1];
    const float* v       = (const float*)d_in[2];
    const float* state   = (const float*)d_in[3];
    const float* A_log   = (const float*)d_in[4];
    const float* a_param = (const float*)d_in[5];
    const float* dt_bias = (const float*)d_in[6];
    const float* b_param = (const float*)d_in[7];

    // tuple output concatenated flat: bf16 out [B,1,HV,D] then f32 new_state [B,HV,D,D]
    __hip_bfloat16* out = (__hip_bfloat16*)d_out;
    float* new_state = (float*)((char*)d_out +
                                (size_t)B_ * HV_ * D_ * sizeof(__hip_bfloat16));

    dim3 grid(B_ * HV_);
    dim3 block(WAVES * 32);
    hipLaunchKernelGGL(gdn_decode_kernel, grid, block, 0, stream,
                       q, k, v, state, A_log, a_param, dt_bias, b_param,
                       out, new_state);
}